// BartPooler_53815940219079
// MI455X (gfx1250) — compile-verified
//
#include <hip/hip_runtime.h>
#include <hip/hip_bf16.h>
#include <math.h>

// Problem constants (match reference):
//   B=16, S=4096, H=1024, T=16, N=B*T=256, K=2H=2048, D_OUT=1024
#define B_  16
#define S_  4096
#define H_  1024
#define T_  16
#define N_  (B_ * T_)      // 256 segments
#define K_  (2 * H_)       // 2048 GEMM reduction dim
#define DO_ 1024           // output features

typedef __attribute__((ext_vector_type(2))) float v2f;
typedef __attribute__((ext_vector_type(8))) float v8f;

// -------- Kernel 1: segment boundaries (trivial: 256 segments) --------------
__global__ __launch_bounds__(N_) void seg_bounds_kernel(
    const int* __restrict__ parts,   // [B, T]
    const int* __restrict__ turns,   // [B]
    int* __restrict__ seg_start,     // [N]
    int* __restrict__ seg_len) {     // [N]
  const int n = threadIdx.x;         // 0..255
  const int b = n >> 4;
  const int t = n & (T_ - 1);
  int start = 1;                     // token 0 is skipped (rel = pos - 1)
  for (int j = 0; j < t; ++j) start += parts[b * T_ + j];
  const int len = (t < turns[b]) ? parts[b * T_ + t] : 0;
  seg_start[n] = start;
  seg_len[n]   = len;
}

// -------- Kernel 2: streaming segment pooling (HBM-bound stage) -------------
// grid = (N, H/256), block = 256. Each thread owns one channel; reads are
// fully coalesced (contiguous 1 KB per 8-wave block per token row). Only
// valid tokens are touched (~half of hidden_states), so this runs at the
// 23.3 TB/s HBM roofline.
__global__ __launch_bounds__(256) void pool_kernel(
    const float* __restrict__ hs,        // [B, S, H]
    const int* __restrict__ seg_start,   // [N]
    const int* __restrict__ seg_len,     // [N]
    float* __restrict__ feat) {          // [N, 2H]  (max | mean)
  const int n = blockIdx.x;                              // segment id
  const int c = blockIdx.y * 256 + threadIdx.x;          // channel 0..1023
  const int b = n >> 4;
  const int start = seg_start[n];
  const int len   = seg_len[n];

  const float* base = hs + ((size_t)b * S_ + (size_t)start) * H_ + c;
  float s  = 0.0f;
  float mx = -INFINITY;
  for (int t = 0; t < len; ++t) {
    // Hint the prefetcher a few rows ahead (emits global_prefetch_b8).
    if (t + 8 < len) __builtin_prefetch(base + (size_t)(t + 8) * H_, 0, 1);
    const float v = base[(size_t)t * H_];
    s += v;
    mx = fmaxf(mx, v);
  }
  float mean;
  if (len > 0) {
    mean = s / (float)len;
  } else {          // cannot happen with reference inputs (parts >= 1)
    mean = 0.0f;
    mx   = 0.0f;
  }
  feat[(size_t)n * K_ + c]      = mx;    // max half
  feat[(size_t)n * K_ + H_ + c] = mean;  // mean half
}

// -------- Kernel 3: GEMM + bias + tanh via V_WMMA_F32_16X16X4_F32 -----------
// out[n, d] = tanh( sum_k feat[n,k] * W[d,k] + bias[d] )
// One wave per 16x16 output tile. A = feat tile (16 rows n x 4 k),
// B = W^T tile (4 k x 16 cols d). ISA 16x4 f32 A layout:
//   lane L: row = L&15; VGPR0 = A[row][2*(L>>4)], VGPR1 = A[row][2*(L>>4)+1]
// B layout is symmetric with cols across lanes. C/D (16x16 f32, 8 VGPRs):
//   lane L, vgpr v -> element [M = v + 8*(L>>4)][Nc = L&15].
__global__ __launch_bounds__(32) void gemm_tanh_kernel(
    const float* __restrict__ feat,   // [N, K]
    const float* __restrict__ W,      // [DO, K]   (row-major: W[d][k])
    const float* __restrict__ bias,   // [DO]
    float* __restrict__ out) {        // [N, DO]
  const int lane = threadIdx.x;          // 0..31
  const int d0 = blockIdx.x * 16;        // output-feature tile
  const int n0 = blockIdx.y * 16;        // segment tile
  const int r  = lane & 15;              // row of A / col of B
  const int g  = lane >> 4;              // K half-group (0 or 1)
  const int kb = g * 2;                  // K offset within 4-wide step

  const float* __restrict__ arow = feat + (size_t)(n0 + r) * K_;
  const float* __restrict__ brow = W    + (size_t)(d0 + r) * K_; // B[k][col]=W[col][k]

  v8f c = {};
#pragma unroll 4
  for (int k = 0; k < K_; k += 4) {
    const v2f a = *(const v2f*)(arow + k + kb);   // 8B aligned (k+kb even)
    const v2f b = *(const v2f*)(brow + k + kb);
    // 8 args: (neg_a, A, neg_b, B, c_mod, C, reuse_a, reuse_b)
    c = __builtin_amdgcn_wmma_f32_16x16x4_f32(
        /*neg_a=*/false, a, /*neg_b=*/false, b,
        /*c_mod=*/(short)0, c, /*reuse_a=*/false, /*reuse_b=*/false);
  }

  const float bv = bias[d0 + r];   // r == N-column of D for this lane
#pragma unroll
  for (int v = 0; v < 8; ++v) {
    const int m = v + 8 * g;
    out[(size_t)(n0 + m) * DO_ + (d0 + r)] = tanhf(c[v] + bv);
  }
}

// ---------------------------------------------------------------------------
extern "C" void kernel_launch(void* const* d_in, const int* in_sizes, int n_in,
                              void* d_out, int out_size, void* d_ws, size_t ws_size,
                              hipStream_t stream) {
  const float* hidden = (const float*)d_in[0];  // [B, S, H]
  const float* W      = (const float*)d_in[1];  // [DO, 2H]
  const float* bias   = (const float*)d_in[2];  // [DO]
  const int*   turns  = (const int*)d_in[3];    // [B]
  const int*   parts  = (const int*)d_in[4];    // [B, T]
  float* out = (float*)d_out;                   // [N, DO]

  // Workspace layout: [seg_start: N ints][seg_len: N ints][pad to 4KB][feat: N*K f32]
  int*   seg_start = (int*)d_ws;
  int*   seg_len   = seg_start + N_;
  float* feat      = (float*)((char*)d_ws + 4096);   // 2 MB

  seg_bounds_kernel<<<1, N_, 0, stream>>>(parts, turns, seg_start, seg_len);

  dim3 pg(N_, H_ / 256);
  pool_kernel<<<pg, 256, 0, stream>>>(hidden, seg_start, seg_len, feat);

  dim3 gg(DO_ / 16, N_ / 16);   // (64, 16) tiles
  gemm_tanh_kernel<<<gg, 32, 0, stream>>>(feat, W, bias, out);
}